// SAGE_45921790329153
// MI455X (gfx1250) — compile-verified
//
#include <hip/hip_runtime.h>
#include <hip/hip_bf16.h>

typedef __attribute__((ext_vector_type(2))) float v2f;
typedef __attribute__((ext_vector_type(8))) float v8f;

#define Hc 128
#define LDS_STRIDE 260   // 260 % 64 == 4 -> conflict-free lane->bank mapping

// ---------------- utility fills ----------------
__global__ void sage_fill0_f4(float4* p, long long n4) {
    long long i = (long long)blockIdx.x * blockDim.x + threadIdx.x;
    if (i < n4) p[i] = make_float4(0.f, 0.f, 0.f, 0.f);
}
__global__ void sage_fill0_f(float* p, int n) {
    int i = blockIdx.x * blockDim.x + threadIdx.x;
    if (i < n) p[i] = 0.f;
}

// ---------------- degree ----------------
__global__ void sage_degree(const int* __restrict__ dst, float* deg, int E_) {
    int e = blockIdx.x * blockDim.x + threadIdx.x;
    if (e < E_) atomicAdd(deg + dst[e], 1.0f);
}
__global__ void sage_invdeg(float* deg, int n) {
    int i = blockIdx.x * blockDim.x + threadIdx.x;
    if (i < n) deg[i] = 1.0f / fmaxf(deg[i], 1.0f);
}

// ---------------- z embedding gather ----------------
__global__ void sage_embed(const int* __restrict__ z, const float* __restrict__ table,
                           float* __restrict__ x, int n) {
    int tid = blockIdx.x * blockDim.x + threadIdx.x;       // n * 32 threads
    if (tid >= n * 32) return;
    int node = tid >> 5;
    int c4   = (tid & 31) << 2;
    int zi = z[node];
    *(float4*)(x + (long long)node * Hc + c4) =
        *(const float4*)(table + (long long)zi * Hc + c4);
}

// ---------------- edge scatter-add (segment_sum of x[src] into summed[dst]) ----
// 32 lanes cover one edge's 128-float row -> fully coalesced 512B row gather,
// f32 atomics resolve in L2 (whole 64MB feature buffer is L2-resident on MI455X).
__global__ __launch_bounds__(256)
void sage_scatter(const int* __restrict__ src, const int* __restrict__ dst,
                  const float* __restrict__ x, float* summed, int E_) {
    long long tid = (long long)blockIdx.x * blockDim.x + threadIdx.x;
    if (tid >= (long long)E_ * 32) return;
    int e  = (int)(tid >> 5);
    int c4 = ((int)tid & 31) << 2;
    int s = src[e], d = dst[e];
    float4 v = *(const float4*)(x + (long long)s * Hc + c4);
    float* p = summed + (long long)d * Hc + c4;
    atomicAdd(p + 0, v.x);
    atomicAdd(p + 1, v.y);
    atomicAdd(p + 2, v.z);
    atomicAdd(p + 3, v.w);
}

// ---------------- fused SAGEConv GEMM via WMMA f32 16x16x4 ----------------
// out = (summed * invdeg) @ Wl + x @ Wr + bl  [+ReLU]
// Treated as one K=256 GEMM: A = [mean | x] (16 x 256 tile in LDS),
// B = [Wl ; Wr] (256 x 128, read from global/L2).
// Block: 256 thr = 8 waves; block -> 16 rows, wave w -> cols [16w,16w+16).
// NOTE: out may alias summed (block reads its rows into LDS before writing).
__global__ __launch_bounds__(256)
void sage_gemm(const float* __restrict__ x, const float* summed,
               const float* __restrict__ invdeg,
               const float* __restrict__ Wl, const float* __restrict__ bl,
               const float* __restrict__ Wr,
               float* out, int relu) {
    __shared__ float As[16 * LDS_STRIDE];
    const int m0  = blockIdx.x * 16;
    const int tid = threadIdx.x;

    // Stage A tile: cols 0..127 = summed*invdeg (mean), cols 128..255 = x
    #pragma unroll
    for (int it = 0; it < 4; ++it) {
        int idx  = tid + it * 256;        // 0..1023 float4 slots (16 rows x 64)
        int row  = idx >> 6;              // 0..15
        int c4   = (idx & 63) << 2;       // 0..252
        int grow = m0 + row;
        float4 v;
        if (c4 < Hc) {
            v = *(const float4*)(summed + (long long)grow * Hc + c4);
            float s = invdeg[grow];
            v.x *= s; v.y *= s; v.z *= s; v.w *= s;
        } else {
            v = *(const float4*)(x + (long long)grow * Hc + (c4 - Hc));
        }
        *(float4*)(&As[row * LDS_STRIDE + c4]) = v;
    }
    __syncthreads();

    const int lane = tid & 31;
    const int wave = tid >> 5;            // 0..7
    const int half = lane >> 4;           // K-pair select per ISA A/B layout
    const int ln   = lane & 15;           // M for A, N for B/C
    const int n0   = wave * 16;

    v8f c = {};
    for (int k0 = 0; k0 < 256; k0 += 4) {
        const float* W = (k0 < Hc) ? Wl : Wr;
        int kk = (k0 & (Hc - 1)) + 2 * half;
        v2f a, b;
        a.x = As[ln * LDS_STRIDE + k0 + 2 * half];
        a.y = As[ln * LDS_STRIDE + k0 + 2 * half + 1];
        b.x = W[kk * Hc + n0 + ln];
        b.y = W[(kk + 1) * Hc + n0 + ln];
        c = __builtin_amdgcn_wmma_f32_16x16x4_f32(
                /*neg_a=*/false, a, /*neg_b=*/false, b,
                /*c_mod=*/(short)0, c, /*reuse_a=*/false, /*reuse_b=*/false);
    }

    float bias = bl[n0 + ln];
    #pragma unroll
    for (int i = 0; i < 8; ++i) {
        float v = c[i] + bias;
        if (relu) v = fmaxf(v, 0.0f);
        out[(long long)(m0 + i + 8 * half) * Hc + (n0 + ln)] = v;
    }
}

// ---------------- per-graph readout MLP ----------------
__global__ __launch_bounds__(128)
void sage_readout(const float* __restrict__ xf, const int* __restrict__ batch,
                  const float* __restrict__ W1, const float* __restrict__ b1,
                  const float* __restrict__ W2, const float* __restrict__ b2,
                  float* __restrict__ out, int Ntot) {
    __shared__ float h[128];
    __shared__ float red[128];
    int g = blockIdx.x;
    int j = threadIdx.x;
    // lower_bound of g in sorted batch[]
    int lo = 0, hi = Ntot;
    while (lo < hi) {
        int mid = (lo + hi) >> 1;
        if (batch[mid] < g) lo = mid + 1; else hi = mid;
    }
    int c = lo;
    h[j] = xf[(long long)c * Hc + j] * xf[(long long)(c + 1) * Hc + j];
    __syncthreads();
    float t = b1[j];
    for (int k = 0; k < Hc; ++k) t = fmaf(h[k], W1[k * Hc + j], t);
    t = fmaxf(t, 0.0f);
    red[j] = t * W2[j];
    __syncthreads();
    for (int s = 64; s > 0; s >>= 1) {
        if (j < s) red[j] += red[j + s];
        __syncthreads();
    }
    if (j == 0) out[g] = red[0] + b2[0];
}

extern "C" void kernel_launch(void* const* d_in, const int* in_sizes, int n_in,
                              void* d_out, int out_size, void* d_ws, size_t ws_size,
                              hipStream_t stream) {
    const int*   z        = (const int*)d_in[0];
    const int*   edge_idx = (const int*)d_in[1];
    const int*   batch    = (const int*)d_in[2];
    const float* z_table  = (const float*)d_in[4];
    const float* Wl       = (const float*)d_in[5];
    const float* bl       = (const float*)d_in[6];
    const float* Wr       = (const float*)d_in[7];
    const float* W1       = (const float*)d_in[8];
    const float* b1       = (const float*)d_in[9];
    const float* W2       = (const float*)d_in[10];
    const float* b2       = (const float*)d_in[11];

    const int N_ = in_sizes[0];
    const int E_ = in_sizes[1] / 2;
    const int G_ = out_size;

    const int* srcI = edge_idx;
    const int* dstI = edge_idx + E_;

    float* buf0 = (float*)d_ws;                       // N*H
    float* buf1 = buf0 + (size_t)N_ * Hc;             // N*H
    float* deg  = buf1 + (size_t)N_ * Hc;             // N (deg -> invdeg in place)

    // degree once (shared by all layers)
    sage_fill0_f<<<(N_ + 255) / 256, 256, 0, stream>>>(deg, N_);
    sage_degree<<<(E_ + 255) / 256, 256, 0, stream>>>(dstI, deg, E_);
    sage_invdeg<<<(N_ + 255) / 256, 256, 0, stream>>>(deg, N_);

    // x0 = z_table[z]
    sage_embed<<<((long long)N_ * 32 + 255) / 256, 256, 0, stream>>>(z, z_table, buf0, N_);

    float* xin = buf0;
    float* xout = buf1;
    const long long feat4 = (long long)N_ * Hc / 4;
    for (int l = 0; l < 3; ++l) {
        sage_fill0_f4<<<(int)((feat4 + 255) / 256), 256, 0, stream>>>((float4*)xout, feat4);
        sage_scatter<<<(int)(((long long)E_ * 32 + 255) / 256), 256, 0, stream>>>(
            srcI, dstI, xin, xout, E_);
        sage_gemm<<<N_ / 16, 256, 0, stream>>>(
            xin, xout, deg,
            Wl + (size_t)l * Hc * Hc, bl + (size_t)l * Hc, Wr + (size_t)l * Hc * Hc,
            xout, (l < 2) ? 1 : 0);
        float* t = xin; xin = xout; xout = t;
    }

    sage_readout<<<G_, 128, 0, stream>>>(xin, batch, W1, b1, W2, b2, (float*)d_out, N_);
}